// CI01FromAntichain_64235530879326
// MI455X (gfx1250) — compile-verified
//
#include <hip/hip_runtime.h>
#include <stdint.h>

#define NFEAT    32
#define NGROUPS  64
#define GSZ      3
#define BLOCK    256
#define RPT      2                    // rows per thread
#define ROWPAD   33                   // LDS dwords per row (TDM pads 1 dword per 32)
#define PAIRSTR  (RPT * ROWPAD)       // 66 dwords per thread: bank = 2l+c, conflict-free
#define TILEROWS (BLOCK * RPT)        // 512 rows per block

typedef __attribute__((ext_vector_type(4))) unsigned int v4u;
typedef __attribute__((ext_vector_type(8))) int          v8i;
typedef __attribute__((ext_vector_type(4))) int          v4i;

__global__ __launch_bounds__(BLOCK, 2)
void antichain_minmax_kernel(const float* __restrict__ x,
                             const int*   __restrict__ groups,
                             float* __restrict__ out,
                             int nrows)
{
    __shared__ float tile[TILEROWS * ROWPAD];   // 66 KB per block

    const int t       = threadIdx.x;
    const int rowBase = blockIdx.x * TILEROWS;

#if __has_builtin(__builtin_amdgcn_tensor_load_to_lds)
    // --- TDM staging: one DMA moves 512 rows x 128B into LDS, hardware-padding
    // --- 1 dword after every 32 dwords -> conflict-free stride-33 rows.
    if (t < 32) {
        unsigned long long gaddr =
            (unsigned long long)(uintptr_t)(x + (size_t)rowBase * NFEAT);
        unsigned lds_base = (unsigned)(uintptr_t)(&tile[0]);  // low 32 bits = LDS byte offset

        const unsigned nelem = TILEROWS * NFEAT;              // 16384 elements (4B each)

        v4u g0;
        g0.x = 1u;                                            // count=1 (valid descriptor)
        g0.y = lds_base;                                      // lds_addr (bytes)
        g0.z = (unsigned)(gaddr & 0xFFFFFFFFull);             // global_addr[31:0]
        g0.w = (unsigned)((gaddr >> 32) & 0x1FFFFFFull)       // global_addr[56:32]
             | (2u << 30);                                    // type = 2 ("image")

        v8i g1 = {0, 0, 0, 0, 0, 0, 0, 0};
        g1[0] = (2 << 16)        // data_size = 4 bytes
              | (1 << 20)        // pad_enable
              | (4 << 22);       // pad_interval = 32 dwords; pad_amount field 0 = 1 dword
        g1[1] = (int)((nelem & 0xFFFFu) << 16);               // tensor_dim0[15:0] -> bits[63:48]
        g1[2] = (int)((nelem >> 16) | (1u << 16));            // tensor_dim0[31:16], tensor_dim1=1
        g1[3] = (int)((nelem & 0xFFFFu) << 16);               // tile_dim0 = 16384 -> bits[127:112]
        g1[4] = 0;                                            // tile_dim1 = tile_dim2 = 0 (unused)
        g1[5] = (int)nelem;                                   // tensor_dim0_stride[31:0]
        g1[6] = 0;
        g1[7] = 0;

        v4i gz4 = {0, 0, 0, 0};
        v8i gz8 = {0, 0, 0, 0, 0, 0, 0, 0};
        __builtin_amdgcn_tensor_load_to_lds(g0, g1, gz4, gz4, gz8, 0);
        __builtin_amdgcn_s_wait_tensorcnt(0);
    }
#else
    // --- Fallback staging (dead on this toolchain): scalar b32 stores, stride 33.
    {
        const int r0 = rowBase + 2 * t;
        if (r0 + 1 < nrows) {
            #pragma unroll
            for (int k = 0; k < NFEAT; ++k) {
                tile[t * PAIRSTR + k]          = x[(size_t)r0 * NFEAT + k];
                tile[t * PAIRSTR + ROWPAD + k] = x[(size_t)(r0 + 1) * NFEAT + k];
            }
        }
    }
#endif
    __syncthreads();

    // --- Gather/min/max for 2 rows per thread. The paired reads at +0/+33 dwords
    // --- share one address VGPR -> ds_load_2addr_b32 (3 DS per group per 2 rows).
    const int tb = t * PAIRSTR;
    float acc0 = -__builtin_huge_valf();
    float acc1 = -__builtin_huge_valf();
    #pragma unroll
    for (int g = 0; g < NGROUPS; ++g) {
        const int i0 = groups[g * GSZ + 0];
        const int i1 = groups[g * GSZ + 1];
        const int i2 = groups[g * GSZ + 2];
        const float a0 = tile[tb + i0];
        const float a1 = tile[tb + ROWPAD + i0];
        const float b0 = tile[tb + i1];
        const float b1 = tile[tb + ROWPAD + i1];
        const float c0 = tile[tb + i2];
        const float c1 = tile[tb + ROWPAD + i2];
        acc0 = fmaxf(acc0, fminf(fminf(a0, b0), c0));   // -> v_min3 + v_max
        acc1 = fmaxf(acc1, fminf(fminf(a1, b1), c1));
    }

    const int r0 = rowBase + 2 * t;
    if (r0 + 1 < nrows) {
        *(float2*)(out + r0) = make_float2(acc0, acc1);  // coalesced global_store_b64
    }
}

extern "C" void kernel_launch(void* const* d_in, const int* in_sizes, int n_in,
                              void* d_out, int out_size, void* d_ws, size_t ws_size,
                              hipStream_t stream) {
    (void)n_in; (void)out_size; (void)d_ws; (void)ws_size;
    const float* x      = (const float*)d_in[0];
    const int*   groups = (const int*)d_in[1];
    float*       out    = (float*)d_out;

    const int nrows   = in_sizes[0] / NFEAT;               // B = 1048576
    const int nblocks = (nrows + TILEROWS - 1) / TILEROWS; // 2048

    antichain_minmax_kernel<<<dim3(nblocks), dim3(BLOCK), 0, stream>>>(x, groups, out, nrows);
}